// DilateLoss_55783035240781
// MI455X (gfx1250) — compile-verified
//
#include <hip/hip_runtime.h>

// DILATE loss (soft-DTW fwd + grad) for MI455X / gfx1250.
// One workgroup (128 threads = 4 wave32) per sequence m (grid = 256).
// R (129x130) and D (128x130) live entirely in LDS (~137 KB/block; CDNA5 WGP
// has 320 KB LDS -> 2 blocks/WGP). D is built with V_WMMA_F32_16X16X4_F32
// via the exact rank-4 decomposition (t-o)^2 = t^2*1 + (-2t)*o + 1*o^2.
// Strides 130 keep anti-diagonal LDS accesses bank-conflict-free
// (addr = i*129 + d, 129 == 1 mod 64).
// WMMA operand construction is branchless (v_cndmask) so EXEC stays all-ones
// and the per-tile LDS loads pipeline under the previous tile's stores.

#define NN      128
#define TT      128
#define NSEQ    256
#define BATCH   64
#define GAMMA_F 0.01f
#define INVG    100.0f
#define ALPHA_F 0.5f
#define BIGV    1.0e8f
#define SD      130
#define SR      130

typedef __attribute__((ext_vector_type(2))) float v2f;
typedef __attribute__((ext_vector_type(8))) float v8f;

__launch_bounds__(128, 2)
__global__ void dilate_softdtw_kernel(const float* __restrict__ outputs,
                                      const float* __restrict__ targets,
                                      float* __restrict__ vals,
                                      float* __restrict__ partials)
{
    __shared__ float tsh[TT];
    __shared__ float osh[TT];
    __shared__ float Dm[NN * SD];            // 66,560 B
    __shared__ float Rm[(NN + 1) * SR];      // 67,080 B
    __shared__ float Ebuf[3 * SR];           // rotating E diagonals
    __shared__ float red[128];

    const int m   = blockIdx.x;              // sequence id, 0..255
    const int b   = m >> 2;
    const int c   = m & 3;
    const int tid = threadIdx.x;             // 0..127
    const int i   = tid + 1;                 // DP row (1-based)

    // t[m][k] = targets[b][k][c] ; o[m][k] = outputs[b][k][c]
    tsh[tid] = targets[(b * TT + tid) * 4 + c];
    osh[tid] = outputs[(b * TT + tid) * 4 + c];
    __syncthreads();

    // ---------------- cost matrix via WMMA f32 16x16x4 ----------------
    // C[M][N] = t^2*1 + (-2t)*o + 1*o^2 + 0*0 = (t - o)^2
    // A lanes 0-15 supply K=0,1 for row M=l; lanes 16-31 supply K=2,3.
    // B lanes 0-15 supply rows K=0,1 at col N=l; lanes 16-31 rows K=2,3.
    {
        const int wave = tid >> 5;
        const int lane = tid & 31;
        const bool h0  = (lane < 16);
        const int l    = lane & 15;          // M (for A) / N (for B)
        for (int k = 0; k < 16; ++k) {       // 64 tiles / 4 waves
            const int tile = wave * 16 + k;
            const int ti = tile >> 3;
            const int tj = tile & 7;
            const float tv = tsh[ti * 16 + l];
            const float ov = osh[tj * 16 + l];
            v2f av, bv;
            av.x = h0 ? tv * tv      : 1.0f;     // A[M][0] / A[M][2]
            av.y = h0 ? -2.0f * tv   : 0.0f;     // A[M][1] / A[M][3]
            bv.x = h0 ? 1.0f         : ov * ov;  // B[0][N] / B[2][N]
            bv.y = h0 ? ov           : 0.0f;     // B[1][N] / B[3][N]
            v8f cv = {};
            cv = __builtin_amdgcn_wmma_f32_16x16x4_f32(
                    false, av, false, bv, (short)0, cv, false, false);
            const int col  = tj * 16 + l;
            const int row0 = ti * 16 + (h0 ? 0 : 8);
            #pragma unroll
            for (int r = 0; r < 8; ++r)
                Dm[(row0 + r) * SD + col] = cv[r];
        }
    }

    // ---------------- forward soft-DTW (anti-diagonal DP) -------------
    Rm[i * SR + 0] = BIGV;
    Rm[0 * SR + i] = BIGV;
    if (tid == 0) Rm[0] = 0.0f;
    __syncthreads();

    for (int d = 2; d <= 2 * NN; ++d) {
        const int j = d - i;
        if (j >= 1 && j <= NN) {
            const float rd = Rm[(i - 1) * SR + (j - 1)];
            const float ru = Rm[(i - 1) * SR + j];
            const float rl = Rm[i * SR + (j - 1)];
            const float mn = fminf(rd, fminf(ru, rl));
            const float s  = __expf((mn - rd) * INVG)
                           + __expf((mn - ru) * INVG)
                           + __expf((mn - rl) * INVG);
            const float sm = mn - GAMMA_F * __logf(s);
            Rm[i * SR + j] = Dm[(i - 1) * SD + (j - 1)] + sm;
        }
        __syncthreads();
    }

    // ---------------- backward: E (gradient of R[N,N] wrt D) ----------
    float* En1  = &Ebuf[0];          // E on diagonal d+1
    float* En2  = &Ebuf[SR];         // E on diagonal d+2
    float* Ecur = &Ebuf[2 * SR];
    for (int idx = tid; idx < 3 * SR; idx += 128) Ebuf[idx] = 0.0f;
    __syncthreads();
    if (tid == NN - 1) En1[NN] = 1.0f;   // E[N][N] = 1 (diag 2N)
    __syncthreads();

    float acc = 0.0f;
    for (int d = 2 * NN - 1; d >= 2; --d) {
        const int j = d - i;
        float E = 0.0f;
        if (j >= 1 && j <= NN) {
            const float Rc = Rm[i * SR + j];
            float aT = 0.0f, bT = 0.0f, cT = 0.0f;
            if (i < NN)            // (i+1, j)  : D 0-based [i][j-1]
                aT = __expf((Rm[(i + 1) * SR + j] - Rc - Dm[i * SD + (j - 1)]) * INVG) * En1[i + 1];
            if (j < NN)            // (i, j+1)  : D 0-based [i-1][j]
                bT = __expf((Rm[i * SR + (j + 1)] - Rc - Dm[(i - 1) * SD + j]) * INVG) * En1[i];
            if (i < NN && j < NN)  // (i+1, j+1): D 0-based [i][j]
                cT = __expf((Rm[(i + 1) * SR + (j + 1)] - Rc - Dm[i * SD + j]) * INVG) * En2[i + 1];
            E = aT + bT + cT;
            const float df = (float)(i - j);
            acc += E * df * df;    // Omega[i-1][j-1] = (i-j)^2
        }
        Ecur[i] = E;
        __syncthreads();
        float* tswap = En2; En2 = En1; En1 = Ecur; Ecur = tswap;
    }

    // ---------------- block reduction of temporal term ----------------
    red[tid] = acc;
    __syncthreads();
    for (int s = 64; s > 0; s >>= 1) {
        if (tid < s) red[tid] += red[tid + s];
        __syncthreads();
    }
    if (tid == 0) {
        partials[m] = red[0];
        vals[m]     = Rm[NN * SR + NN];
    }
}

__launch_bounds__(256, 1)
__global__ void dilate_finalize_kernel(const float* __restrict__ vals,
                                       const float* __restrict__ partials,
                                       float* __restrict__ out)
{
    __shared__ float red[256];
    const int tid = threadIdx.x;
    const float wShape = ALPHA_F / (float)BATCH;
    const float wTemp  = (1.0f - ALPHA_F) / ((float)BATCH * (float)TT * (float)TT);
    red[tid] = wShape * vals[tid] + wTemp * partials[tid];
    __syncthreads();
    for (int s = 128; s > 0; s >>= 1) {
        if (tid < s) red[tid] += red[tid + s];
        __syncthreads();
    }
    if (tid == 0) out[0] = red[0];
}

extern "C" void kernel_launch(void* const* d_in, const int* in_sizes, int n_in,
                              void* d_out, int out_size, void* d_ws, size_t ws_size,
                              hipStream_t stream)
{
    const float* outputs = (const float*)d_in[0];  // [64,128,4] f32
    const float* targets = (const float*)d_in[1];  // [64,128,4] f32
    float* vals     = (float*)d_ws;                // 256 floats
    float* partials = vals + NSEQ;                 // 256 floats

    dilate_softdtw_kernel<<<NSEQ, 128, 0, stream>>>(outputs, targets, vals, partials);
    dilate_finalize_kernel<<<1, 256, 0, stream>>>(vals, partials, (float*)d_out);
}